// PointNetSetAbstraction_4097398800755
// MI455X (gfx1250) — compile-verified
//
#include <hip/hip_runtime.h>
#include <hip/hip_bf16.h>

typedef __attribute__((ext_vector_type(2))) float v2f;
typedef __attribute__((ext_vector_type(8))) float v8f;

#define BB    8
#define NN    4096
#define SS    1024      // NPOINT
#define NS    32        // NSAMPLE
#define INC   64
#define K0    70
#define K0P   72
#define C0    64
#define C1    64
#define C2    128
#define MM    (BB*SS*NS)   // 262144
#define R2    0.25f
#define EPSV  1e-5f

// ---------------------------------------------------------------------------
// 1) Farthest point sampling: one 1024-thread workgroup per batch.
//    Each thread owns 4 points in registers; centroid broadcast via LDS.
// ---------------------------------------------------------------------------
__global__ void fps_kernel(const float* __restrict__ xyz, float* __restrict__ new_xyz) {
  const int b = blockIdx.x;
  const int t = threadIdx.x;                 // 0..1023
  const float* base = xyz + (size_t)b * NN * 3;

  float px[4], py[4], pz[4], dist[4];
  for (int p = 0; p < 4; ++p) {
    int i = t + p * 1024;
    px[p] = base[3*i + 0];
    py[p] = base[3*i + 1];
    pz[p] = base[3*i + 2];
    dist[p] = 1e10f;
  }

  __shared__ float cent[3];
  __shared__ float rv[32];
  __shared__ int   ri[32];

  int far = 0;
  for (int s = 0; s < SS; ++s) {
    if (t == (far & 1023)) {
      int p = far >> 10;
      cent[0] = px[p]; cent[1] = py[p]; cent[2] = pz[p];
    }
    __syncthreads();
    float cx = cent[0], cy = cent[1], cz = cent[2];
    if (t < 3) new_xyz[((size_t)b * SS + s) * 3 + t] = cent[t];

    float bestv = -1.f; int besti = 0;
    for (int p = 0; p < 4; ++p) {
      int i = t + p * 1024;
      float dx = px[p]-cx, dy = py[p]-cy, dz = pz[p]-cz;
      float d = dx*dx + dy*dy + dz*dz;
      float m = fminf(dist[p], d);
      dist[p] = m;
      if (m > bestv) { bestv = m; besti = i; }   // ascending i -> first-index tie
    }
    // wave32 argmax reduction
    for (int off = 16; off; off >>= 1) {
      float ov = __shfl_xor(bestv, off, 32);
      int   oi = __shfl_xor(besti, off, 32);
      if (ov > bestv || (ov == bestv && oi < besti)) { bestv = ov; besti = oi; }
    }
    if ((t & 31) == 0) { rv[t >> 5] = bestv; ri[t >> 5] = besti; }
    __syncthreads();
    if (t < 32) {
      bestv = rv[t]; besti = ri[t];
      for (int off = 16; off; off >>= 1) {
        float ov = __shfl_xor(bestv, off, 32);
        int   oi = __shfl_xor(besti, off, 32);
        if (ov > bestv || (ov == bestv && oi < besti)) { bestv = ov; besti = oi; }
      }
      if (t == 0) ri[0] = besti;
    }
    __syncthreads();
    far = ri[0];
    // cent-publish __syncthreads at loop top separates this read from next rv/ri write
  }
}

// ---------------------------------------------------------------------------
// 2) Ball query: one wave32 per centroid; first NS in-radius points in index
//    order via ballot+popcount prefix; pad with first index.
// ---------------------------------------------------------------------------
__global__ void ball_query_kernel(const float* __restrict__ xyz,
                                  const float* __restrict__ new_xyz,
                                  int* __restrict__ idx) {
  int gwave = (int)((blockIdx.x * (size_t)blockDim.x + threadIdx.x) >> 5);
  int lane  = threadIdx.x & 31;
  int b = gwave / SS, s = gwave % SS;

  const float* nz = new_xyz + ((size_t)b * SS + s) * 3;
  float cx = nz[0], cy = nz[1], cz = nz[2];
  const float* base = xyz + (size_t)b * NN * 3;
  int* out = idx + ((size_t)b * SS + s) * NS;

  int cnt = 0, first = -1;
  for (int j0 = 0; j0 < NN && cnt < NS; j0 += 32) {
    int j = j0 + lane;
    float dx = base[3*j+0]-cx, dy = base[3*j+1]-cy, dz = base[3*j+2]-cz;
    float d = dx*dx + dy*dy + dz*dz;
    bool in = (d <= R2);
    unsigned mask = __builtin_amdgcn_ballot_w32(in);
    if (first < 0 && mask) first = j0 + (__ffs(mask) - 1);
    int pos = cnt + __popc(mask & ((1u << lane) - 1u));
    if (in && pos < NS) out[pos] = j;
    cnt += __popc(mask);
  }
  if (cnt < NS) {
    int p = cnt + lane;
    if (p < NS) out[p] = first;
  }
}

// ---------------------------------------------------------------------------
// 3) Build feature matrix (M x 72): [gxyz(3) | gxyz-center(3) | points(64) | pad(2)]
// ---------------------------------------------------------------------------
__global__ void build_feat_kernel(const float* __restrict__ xyz,
                                  const float* __restrict__ points,
                                  const float* __restrict__ new_xyz,
                                  const int* __restrict__ idx,
                                  float* __restrict__ feat) {
  size_t t = (size_t)blockIdx.x * 256 + threadIdx.x;   // over MM*K0P exactly
  int i = (int)(t % K0P);
  size_t m = t / K0P;
  int b = (int)(m / (SS * NS));
  int r = (int)(m % (SS * NS));
  int s = r / NS;
  int pid = idx[m];
  float v;
  if (i < 3)       v = xyz[((size_t)b * NN + pid) * 3 + i];
  else if (i < 6)  v = xyz[((size_t)b * NN + pid) * 3 + (i-3)]
                     - new_xyz[((size_t)b * SS + s) * 3 + (i-3)];
  else if (i < 70) v = points[((size_t)b * NN + pid) * INC + (i-6)];
  else             v = 0.f;
  feat[t] = v;
}

// ---------------------------------------------------------------------------
// 4) Pad weights (o, Kin) -> (o, Kpad) with zeros
// ---------------------------------------------------------------------------
__global__ void pad_w_kernel(const float* __restrict__ w, float* __restrict__ wp,
                             int O, int Kin, int Kpad) {
  int t = blockIdx.x * 256 + threadIdx.x;
  if (t >= O * Kpad) return;
  int o = t / Kpad, i = t % Kpad;
  wp[t] = (i < Kin) ? w[o * Kin + i] : 0.f;
}

__global__ void zero_stats_kernel(float* __restrict__ stats) {
  stats[threadIdx.x] = 0.f;   // 256 floats
}

// ---------------------------------------------------------------------------
// 5) GEMM via V_WMMA_F32_16X16X4_F32.
//    One wave computes 16 rows x FULL Nout (NTN = Nout/16 accumulator tiles):
//      - A fragment loaded from global exactly ONCE per element (no N-tile
//        redundancy; saves ~4-8x L2 activation traffic).
//      - Weights staged once per block into LDS with a padded row stride
//        (KPAD+2 floats -> bank-conflict-free ds_load_b64 B fragments).
//    Per-channel sum/sumsq accumulated with f32 global atomics for BN.
//    Fragment layouts per ISA 7.12.2: A lane r=lane%16 row, half h=lane/16
//    K-pair; B symmetric; D VGPR j -> row m0+j+8h, col = 16q + r.
// ---------------------------------------------------------------------------
template<int KPAD, int NOUT>
__global__ void gemm_stats_kernel(const float* __restrict__ A,
                                  const float* __restrict__ Wp,
                                  const float* __restrict__ bias,
                                  float* __restrict__ Y,
                                  float* __restrict__ stats) {
  constexpr int NTN = NOUT / 16;
  constexpr int LSTR = KPAD + 2;            // padded row stride (even -> 8B aligned)
  __shared__ float sw[NOUT * LSTR];

  // Cooperative stage of the (tiny) weight matrix into LDS
  for (int t = threadIdx.x; t < NOUT * KPAD; t += 256) {
    int o = t / KPAD, i = t % KPAD;
    sw[o * LSTR + i] = Wp[t];
  }
  __syncthreads();

  const int lane = threadIdx.x & 31;
  const int wave = threadIdx.x >> 5;
  const int mt   = blockIdx.x * 8 + wave;   // exact division -> EXEC all-1s
  const int m0   = mt << 4;
  const int r = lane & 15, h = lane >> 4;

  const float* arow = A + (size_t)(m0 + r) * KPAD + 2 * h;
  const float* brow = &sw[(size_t)r * LSTR + 2 * h];

  v8f c[NTN];
#pragma unroll
  for (int q = 0; q < NTN; ++q) c[q] = (v8f){};

#pragma unroll
  for (int kk = 0; kk < KPAD; kk += 4) {
    v2f a = *(const v2f*)(arow + kk);
#pragma unroll
    for (int q = 0; q < NTN; ++q) {
      v2f bfr = *(const v2f*)(brow + q * 16 * LSTR + kk);
      c[q] = __builtin_amdgcn_wmma_f32_16x16x4_f32(false, a, false, bfr,
                                                   (short)0, c[q], false, false);
    }
  }

#pragma unroll
  for (int q = 0; q < NTN; ++q) {
    int col = q * 16 + r;
    float bval = bias[col];
    float ps = 0.f, pss = 0.f;
#pragma unroll
    for (int j = 0; j < 8; ++j) {
      float y = c[q][j] + bval;
      Y[(size_t)(m0 + j + 8 * h) * NOUT + col] = y;
      ps += y; pss += y * y;
    }
    ps  += __shfl_xor(ps, 16, 32);
    pss += __shfl_xor(pss, 16, 32);
    if (lane < 16) {
      atomicAdd(stats + col, ps);
      atomicAdd(stats + NOUT + col, pss);
    }
  }
}

// ---------------------------------------------------------------------------
// 6) BatchNorm (population stats over all M) + ReLU, elementwise
// ---------------------------------------------------------------------------
__global__ void bn_relu_kernel(const float* __restrict__ src, float* __restrict__ dst,
                               const float* __restrict__ stats,
                               const float* __restrict__ gamma,
                               const float* __restrict__ beta, int Nout) {
  size_t t = (size_t)blockIdx.x * 256 + threadIdx.x;   // over MM*Nout exactly
  int o = (int)(t % Nout);
  float mean = stats[o] * (1.f / MM);
  float var  = stats[Nout + o] * (1.f / MM) - mean * mean;
  float v = (src[t] - mean) * rsqrtf(var + EPSV) * gamma[o] + beta[o];
  dst[t] = fmaxf(v, 0.f);
}

// ---------------------------------------------------------------------------
// 7) Final: BN + ReLU + max over the NS samples -> new_points in d_out
// ---------------------------------------------------------------------------
__global__ void bn_relu_max_kernel(const float* __restrict__ Y2,
                                   const float* __restrict__ stats,
                                   const float* __restrict__ gamma,
                                   const float* __restrict__ beta,
                                   float* __restrict__ out) {
  int t = blockIdx.x * 256 + threadIdx.x;   // over BB*SS*C2 exactly
  int o = t % C2, bs = t / C2;
  float mean = stats[o] * (1.f / MM);
  float inv  = rsqrtf(stats[C2 + o] * (1.f / MM) - mean * mean + EPSV);
  float g = gamma[o], be = beta[o];
  const float* base = Y2 + (size_t)bs * NS * C2 + o;
  float mx = 0.f;  // all candidates are post-ReLU (>= 0)
  for (int k = 0; k < NS; ++k) {
    float v = (base[(size_t)k * C2] - mean) * inv * g + be;
    mx = fmaxf(mx, fmaxf(v, 0.f));
  }
  out[(size_t)BB * SS * 3 + (size_t)bs * C2 + o] = mx;
}

// ---------------------------------------------------------------------------
extern "C" void kernel_launch(void* const* d_in, const int* in_sizes, int n_in,
                              void* d_out, int out_size, void* d_ws, size_t ws_size,
                              hipStream_t stream) {
  const float* xyz    = (const float*)d_in[0];
  const float* points = (const float*)d_in[1];
  const float* w0 = (const float*)d_in[2];
  const float* b0 = (const float*)d_in[3];
  const float* g0 = (const float*)d_in[4];
  const float* e0 = (const float*)d_in[5];
  const float* w1 = (const float*)d_in[6];
  const float* b1 = (const float*)d_in[7];
  const float* g1 = (const float*)d_in[8];
  const float* e1 = (const float*)d_in[9];
  const float* w2 = (const float*)d_in[10];
  const float* b2 = (const float*)d_in[11];
  const float* g2 = (const float*)d_in[12];
  const float* e2 = (const float*)d_in[13];
  float* out = (float*)d_out;
  char* ws = (char*)d_ws;

  size_t off = 0;
  auto alloc = [&](size_t bytes) { size_t o = off; off = (off + bytes + 255) & ~(size_t)255; return o; };
  int*   idx   = (int*)  (ws + alloc((size_t)MM * 4));          //   1 MB
  float* stats = (float*)(ws + alloc(256 * 4));                 //   1 KB
  float* wp0   = (float*)(ws + alloc((size_t)C0 * K0P * 4));
  float* wp1   = (float*)(ws + alloc((size_t)C1 * C0  * 4));
  float* wp2   = (float*)(ws + alloc((size_t)C2 * C1  * 4));
  float* featA = (float*)(ws + alloc((size_t)MM * K0P * 4));    //  75.5 MB (reused for Y1)
  float* bufB  = (float*)(ws + alloc((size_t)MM * C2  * 4));    // 134 MB (Y0 then Y2)
  (void)ws_size; (void)in_sizes; (void)n_in; (void)out_size;

  // Stage 1: FPS -> new_xyz directly into d_out[0 .. B*S*3)
  fps_kernel<<<BB, 1024, 0, stream>>>(xyz, out);

  // Stage 2: ball query (8192 waves, 8 waves/block)
  ball_query_kernel<<<(BB * SS) / 8, 256, 0, stream>>>(xyz, out, idx);

  // Stage 3: gather + concat features (M x 72)
  build_feat_kernel<<<(MM * K0P) / 256, 256, 0, stream>>>(xyz, points, out, idx, featA);

  const int gemm_blocks = (MM / 16) / 8;   // 2048: 8 waves/block, one 16-row stripe each

  // Layer 0: 70(->72) -> 64
  pad_w_kernel<<<(C0 * K0P + 255) / 256, 256, 0, stream>>>(w0, wp0, C0, K0, K0P);
  zero_stats_kernel<<<1, 256, 0, stream>>>(stats);
  gemm_stats_kernel<K0P, C0><<<gemm_blocks, 256, 0, stream>>>(featA, wp0, b0, bufB, stats);
  bn_relu_kernel<<<(MM * C0) / 256, 256, 0, stream>>>(bufB, bufB, stats, g0, e0, C0);

  // Layer 1: 64 -> 64 (output into featA region)
  pad_w_kernel<<<(C1 * C0 + 255) / 256, 256, 0, stream>>>(w1, wp1, C1, C0, C0);
  zero_stats_kernel<<<1, 256, 0, stream>>>(stats);
  gemm_stats_kernel<C0, C1><<<gemm_blocks, 256, 0, stream>>>(bufB, wp1, b1, featA, stats);
  bn_relu_kernel<<<(MM * C1) / 256, 256, 0, stream>>>(featA, featA, stats, g1, e1, C1);

  // Layer 2: 64 -> 128 (into bufB), then BN+ReLU+max fused into output
  pad_w_kernel<<<(C2 * C1 + 255) / 256, 256, 0, stream>>>(w2, wp2, C2, C1, C1);
  zero_stats_kernel<<<1, 256, 0, stream>>>(stats);
  gemm_stats_kernel<C1, C2><<<gemm_blocks, 256, 0, stream>>>(featA, wp2, b2, bufB, stats);
  bn_relu_max_kernel<<<(BB * SS * C2) / 256, 256, 0, stream>>>(bufB, stats, g2, e2, out);
}